// GCN_40020505264140
// MI455X (gfx1250) — compile-verified
//
#include <hip/hip_runtime.h>

#define N_NODES 50000
#define N_EDGES 800000
#define IN_DIM 128
#define OUT_DIM 64
#define BN_EPS 1e-5f
#define SLOPE 0.01f

typedef __attribute__((ext_vector_type(2))) float v2f;
typedef __attribute__((ext_vector_type(8))) float v8f;

// ---------------- degrees ----------------
__global__ void deg_accum(const int* __restrict__ src, const int* __restrict__ dst,
                          float* __restrict__ deg_out, float* __restrict__ deg_in) {
    int e = blockIdx.x * blockDim.x + threadIdx.x;
    if (e < N_EDGES) {
        atomicAdd(&deg_out[src[e]], 1.0f);
        atomicAdd(&deg_in[dst[e]], 1.0f);
    }
}

__global__ void deg_rsqrt(float* __restrict__ deg_out, float* __restrict__ deg_in) {
    int i = blockIdx.x * blockDim.x + threadIdx.x;
    if (i < N_NODES) {
        deg_out[i] = rsqrtf(fmaxf(deg_out[i], 1.0f));
        deg_in[i]  = rsqrtf(fmaxf(deg_in[i], 1.0f));
    }
}

// ---------------- conv1: scatter-add of weighted, out-normalized features ----------------
__global__ void conv1_edge(const float* __restrict__ feat, const float* __restrict__ ew,
                           const int* __restrict__ src, const int* __restrict__ dst,
                           const float* __restrict__ rs_out, float* __restrict__ agg) {
    long long i = (long long)blockIdx.x * blockDim.x + threadIdx.x;
    if (i >= (long long)N_EDGES * IN_DIM) return;
    int e = (int)(i >> 7);
    int d = (int)(i & (IN_DIM - 1));
    int s = src[e];
    float v = feat[(long long)s * IN_DIM + d] * rs_out[s] * ew[e];
    atomicAdd(&agg[(long long)dst[e] * IN_DIM + d], v);
}

__global__ void conv1_fin(float* __restrict__ h, const float* __restrict__ rs_in,
                          const float* __restrict__ b1) {
    int i = blockIdx.x * blockDim.x + threadIdx.x;
    if (i < N_NODES * IN_DIM) {
        int r = i >> 7, c = i & (IN_DIM - 1);
        h[i] = h[i] * rs_in[r] + b1[c];
    }
}

// ---------------- batchnorm statistics (per-column over rows) ----------------
__global__ void bn_stats(const float* __restrict__ x, int nrows, int ncols,
                         float* __restrict__ sum, float* __restrict__ sumsq) {
    int c = threadIdx.x;               // blockDim.x == ncols
    float s = 0.f, s2 = 0.f;
    for (int r = blockIdx.x; r < nrows; r += gridDim.x) {
        float v = x[(long long)r * ncols + c];
        s += v; s2 += v * v;
    }
    atomicAdd(&sum[c], s);
    atomicAdd(&sumsq[c], s2);
}

__global__ void bn_fin(const float* __restrict__ sum, const float* __restrict__ sumsq,
                       int ncols, float inv_n, float* __restrict__ mean,
                       float* __restrict__ rstd) {
    int c = blockIdx.x * blockDim.x + threadIdx.x;
    if (c < ncols) {
        float m = sum[c] * inv_n;
        float v = sumsq[c] * inv_n - m * m;
        mean[c] = m;
        rstd[c] = rsqrtf(v + BN_EPS);
    }
}

// ---------------- BN1 + LeakyReLU + fold rsqrt(deg_out) for conv2 GEMM input ----------------
__global__ void bn1_apply(float* __restrict__ h, const float* __restrict__ mean,
                          const float* __restrict__ rstd, const float* __restrict__ gamma,
                          const float* __restrict__ beta, const float* __restrict__ rs_out) {
    int i = blockIdx.x * blockDim.x + threadIdx.x;
    if (i < N_NODES * IN_DIM) {
        int r = i >> 7, c = i & (IN_DIM - 1);
        float v = gamma[c] * (h[i] - mean[c]) * rstd[c] + beta[c];
        v = v > 0.f ? v : SLOPE * v;
        h[i] = v * rs_out[r];
    }
}

// ---------------- fp32 WMMA GEMM: [50000,128] x [128,64] -> [50000,64] ----------------
// One wave computes one 16-row strip x all 64 cols (4 accumulators).
// A layout (16x4 f32): lanes 0-15 M=0..15; VGPR0 = K0 (lo half) / K2 (hi half), VGPR1 = K1/K3.
// B layout (4x16): same K split across halves, N = lane%16.
// C/D: VGPR r -> (M = r + 8*half, N = lane%16).
// W is staged into LDS pre-swizzled as [k/2][n][2] so every B operand
// (W[k][n], W[k+1][n]) is one contiguous aligned 8-byte chunk -> single
// ds_load_b64 directly into the even VGPR pair WMMA wants (no repack movs).
__global__ __launch_bounds__(256)
void gemm_wmma(const float* __restrict__ xs, const float* __restrict__ W,
               float* __restrict__ g) {
    __shared__ float sW[IN_DIM * OUT_DIM];   // 32 KB, packed pair layout
    for (int i = threadIdx.x; i < IN_DIM * OUT_DIM; i += 256) {
        int k = i >> 6;          // W row (0..127)
        int n = i & (OUT_DIM - 1);
        sW[(k >> 1) * (2 * OUT_DIM) + 2 * n + (k & 1)] = W[i];
    }
    __syncthreads();

    int wave = (blockIdx.x * 256 + threadIdx.x) >> 5;
    int lane = threadIdx.x & 31;
    if (wave >= N_NODES / 16) return;        // wave-uniform: EXEC stays all-1 inside

    int m0   = wave * 16;
    int half = lane >> 4;
    int ln   = lane & 15;
    const float* xrow = xs + (long long)(m0 + ln) * IN_DIM;

    v8f acc0 = {}, acc1 = {}, acc2 = {}, acc3 = {};
    for (int k0 = 0; k0 < IN_DIM; k0 += 4) {
        int ka = k0 + 2 * half;              // always even
        v2f a = *(const v2f*)(xrow + ka);
        const float* wp = sW + (ka >> 1) * (2 * OUT_DIM);
        v2f b0 = *(const v2f*)(wp + 2 * ln);
        v2f b1 = *(const v2f*)(wp + 2 * (16 + ln));
        v2f b2 = *(const v2f*)(wp + 2 * (32 + ln));
        v2f b3 = *(const v2f*)(wp + 2 * (48 + ln));
        acc0 = __builtin_amdgcn_wmma_f32_16x16x4_f32(false, a, false, b0, (short)0, acc0, false, false);
        acc1 = __builtin_amdgcn_wmma_f32_16x16x4_f32(false, a, false, b1, (short)0, acc1, false, false);
        acc2 = __builtin_amdgcn_wmma_f32_16x16x4_f32(false, a, false, b2, (short)0, acc2, false, false);
        acc3 = __builtin_amdgcn_wmma_f32_16x16x4_f32(false, a, false, b3, (short)0, acc3, false, false);
    }

#pragma unroll
    for (int r = 0; r < 8; ++r) {
        int orow = m0 + r + 8 * half;
        float* go = g + (long long)orow * OUT_DIM + ln;
        go[0]  = acc0[r];
        go[16] = acc1[r];
        go[32] = acc2[r];
        go[48] = acc3[r];
    }
}

// ---------------- conv2: scatter-add (no edge weight; rs_out already folded in) ----------------
__global__ void conv2_edge(const float* __restrict__ g, const int* __restrict__ src,
                           const int* __restrict__ dst, float* __restrict__ agg) {
    long long i = (long long)blockIdx.x * blockDim.x + threadIdx.x;
    if (i >= (long long)N_EDGES * OUT_DIM) return;
    int e = (int)(i >> 6);
    int d = (int)(i & (OUT_DIM - 1));
    atomicAdd(&agg[(long long)dst[e] * OUT_DIM + d], g[(long long)src[e] * OUT_DIM + d]);
}

__global__ void conv2_fin(float* __restrict__ out, const float* __restrict__ rs_in,
                          const float* __restrict__ b2) {
    int i = blockIdx.x * blockDim.x + threadIdx.x;
    if (i < N_NODES * OUT_DIM) {
        int r = i >> 6, c = i & (OUT_DIM - 1);
        out[i] = out[i] * rs_in[r] + b2[c];
    }
}

// ---------------- BN2 + row softmax (one wave32 per row, 2 cols per lane) ----------------
__global__ void bn2_softmax(float* __restrict__ out, const float* __restrict__ mean,
                            const float* __restrict__ rstd, const float* __restrict__ gamma,
                            const float* __restrict__ beta) {
    int wave = (blockIdx.x * blockDim.x + threadIdx.x) >> 5;
    int lane = threadIdx.x & 31;
    if (wave >= N_NODES) return;
    float* row = out + (long long)wave * OUT_DIM;
    int c0 = lane, c1 = lane + 32;
    float v0 = gamma[c0] * (row[c0] - mean[c0]) * rstd[c0] + beta[c0];
    float v1 = gamma[c1] * (row[c1] - mean[c1]) * rstd[c1] + beta[c1];
    float m = fmaxf(v0, v1);
    for (int off = 16; off > 0; off >>= 1) m = fmaxf(m, __shfl_xor(m, off, 32));
    float e0 = __expf(v0 - m), e1 = __expf(v1 - m);
    float s = e0 + e1;
    for (int off = 16; off > 0; off >>= 1) s += __shfl_xor(s, off, 32);
    float inv = 1.0f / s;
    row[c0] = e0 * inv;
    row[c1] = e1 * inv;
}

extern "C" void kernel_launch(void* const* d_in, const int* in_sizes, int n_in,
                              void* d_out, int out_size, void* d_ws, size_t ws_size,
                              hipStream_t stream) {
    const float* feature     = (const float*)d_in[0];
    const float* edge_weight = (const float*)d_in[1];
    const float* b1          = (const float*)d_in[2];
    const float* gamma1      = (const float*)d_in[3];
    const float* beta1       = (const float*)d_in[4];
    const float* W2          = (const float*)d_in[5];
    const float* b2          = (const float*)d_in[6];
    const float* gamma2      = (const float*)d_in[7];
    const float* beta2       = (const float*)d_in[8];
    const int*   src         = (const int*)d_in[9];
    const int*   dst         = (const int*)d_in[10];
    float* out = (float*)d_out;

    float* ws      = (float*)d_ws;
    float* h       = ws;                                  // N*128
    float* g       = h + (size_t)N_NODES * IN_DIM;        // N*64
    float* rs_out  = g + (size_t)N_NODES * OUT_DIM;       // N
    float* rs_in   = rs_out + N_NODES;                    // N
    float* sum1    = rs_in + N_NODES;                     // 128
    float* sumsq1  = sum1 + IN_DIM;                       // 128
    float* mean1   = sumsq1 + IN_DIM;                     // 128
    float* rstd1   = mean1 + IN_DIM;                      // 128
    float* sum2    = rstd1 + IN_DIM;                      // 64
    float* sumsq2  = sum2 + OUT_DIM;                      // 64
    float* mean2   = sumsq2 + OUT_DIM;                    // 64
    float* rstd2   = mean2 + OUT_DIM;                     // 64

    hipMemsetAsync(h, 0, sizeof(float) * (size_t)N_NODES * IN_DIM, stream);
    hipMemsetAsync(rs_out, 0, sizeof(float) * (size_t)N_NODES * 2, stream);
    hipMemsetAsync(sum1, 0, sizeof(float) * 2 * IN_DIM, stream);
    hipMemsetAsync(sum2, 0, sizeof(float) * 2 * OUT_DIM, stream);
    hipMemsetAsync(out, 0, sizeof(float) * (size_t)N_NODES * OUT_DIM, stream);

    deg_accum<<<(N_EDGES + 255) / 256, 256, 0, stream>>>(src, dst, rs_out, rs_in);
    deg_rsqrt<<<(N_NODES + 255) / 256, 256, 0, stream>>>(rs_out, rs_in);

    conv1_edge<<<(int)(((long long)N_EDGES * IN_DIM) / 256), 256, 0, stream>>>(
        feature, edge_weight, src, dst, rs_out, h);
    conv1_fin<<<(N_NODES * IN_DIM + 255) / 256, 256, 0, stream>>>(h, rs_in, b1);

    bn_stats<<<512, IN_DIM, 0, stream>>>(h, N_NODES, IN_DIM, sum1, sumsq1);
    bn_fin<<<1, IN_DIM, 0, stream>>>(sum1, sumsq1, IN_DIM, 1.0f / N_NODES, mean1, rstd1);
    bn1_apply<<<(N_NODES * IN_DIM + 255) / 256, 256, 0, stream>>>(h, mean1, rstd1,
                                                                  gamma1, beta1, rs_out);

    gemm_wmma<<<(N_NODES / 16 + 7) / 8, 256, 0, stream>>>(h, W2, g);

    conv2_edge<<<(int)(((long long)N_EDGES * OUT_DIM) / 256), 256, 0, stream>>>(g, src, dst, out);
    conv2_fin<<<(N_NODES * OUT_DIM + 255) / 256, 256, 0, stream>>>(out, rs_in, b2);

    bn_stats<<<512, OUT_DIM, 0, stream>>>(out, N_NODES, OUT_DIM, sum2, sumsq2);
    bn_fin<<<1, OUT_DIM, 0, stream>>>(sum2, sumsq2, OUT_DIM, 1.0f / N_NODES, mean2, rstd2);
    bn2_softmax<<<(N_NODES * 32 + 255) / 256, 256, 0, stream>>>(out, mean2, rstd2,
                                                                gamma2, beta2);
}